// _Minibatch_block_72988674228272
// MI455X (gfx1250) — compile-verified
//
#include <hip/hip_runtime.h>
#include <hip/hip_bf16.h>

// ---------------------------------------------------------------------------
// MI455X / gfx1250 implementation.
//
// ~5 MB of traffic, 0.26 GFLOP GEMM -> latency-bound at 23.3 TB/s; fits in a
// sliver of the 192 MB L2. Precision downgrade buys nothing, so the GEMM uses
// the CDNA5 fp32 matrix op V_WMMA_F32_16X16X4_F32 (wave32, 16x16 tile/wave).
// The K-loop is branch-free (EXEC all-1s, as WMMA requires): out-of-range
// B columns are clamped, garbage lanes are zeroed only in the epilogue store.
// LDS staging in the pairwise kernel uses the gfx1250 async-to-LDS path.
// ---------------------------------------------------------------------------

typedef __attribute__((ext_vector_type(2))) float v2f;
typedef __attribute__((ext_vector_type(8))) float v8f;

#define B_SZ   256
#define F_SZ   1024
#define NCOLS  500      // N_KERNELS * DIM_PER_KERNEL
#define NPAD   512      // padded X stride (32 N-tiles of 16)
#define OUTW   1224     // 1024 + 100 + 100
#define INV_DEN (1.0f / 32640.0f)   // 256*128 - 128 self-pairs

#if defined(__HIP_DEVICE_COMPILE__) && \
    __has_builtin(__builtin_amdgcn_global_load_async_to_lds_b32) && \
    __has_builtin(__builtin_amdgcn_s_wait_asynccnt)
#define USE_ASYNC_LDS 1
typedef __attribute__((address_space(1))) int glb_i32;   // global int*
typedef __attribute__((address_space(3))) int lds_i32;   // LDS int*
#endif

// ---------------------------------------------------------------------------
// Kernel 1: X[256][512] = inputs(256x1024) @ W(1024x500) + b, fp32 WMMA.
// One wave per 16x16 tile. Grid: (4 blocks x 8 waves over 32 N-tiles) x
// (16 M-tiles). Fragment layout (ISA 7.12.2, 32-bit 16x4 A / 4x16 B):
//   lane l in [0,16): row/col = l, VGPR0/1 = K {0,1};  lanes 16..31: K {2,3}.
// ---------------------------------------------------------------------------
__global__ __launch_bounds__(256) void gemm_wmma_f32(
    const float* __restrict__ A,     // 256 x 1024 row-major
    const float* __restrict__ W,     // 1024 x 500 row-major
    const float* __restrict__ bias,  // 500
    float*       __restrict__ X)     // 256 x 512 (cols 500..511 zero)
{
    const int lane = threadIdx.x & 31;
    const int wave = threadIdx.x >> 5;
    const int l    = lane & 15;       // row of A-frag / col of B-frag
    const int h    = lane >> 4;       // half-wave: K offset 0 or 2
    const int m0   = blockIdx.y * 16;
    const int n0   = (blockIdx.x * 8 + wave) * 16;
    const int col  = n0 + l;
    const bool colOK = (col < NCOLS);
    // Clamp instead of guarding in-loop: keeps EXEC all-1s and the K-loop
    // branch-free. Clamped lanes produce garbage that is never stored.
    const int colC = colOK ? col : (NCOLS - 1);

    const float* ap = A + (m0 + l) * F_SZ + 2 * h;      // A row, K base 2h
    const float* wp = W + 2 * h * NCOLS + colC;         // B col, K base 2h

    v8f acc = {};
#pragma unroll 8
    for (int k = 0; k < F_SZ; k += 4) {
        if ((k & 127) == 0 && k + 128 < F_SZ)
            __builtin_prefetch(ap + k + 128, 0, 1);     // global_prefetch_b8

        v2f a = *(const v2f*)(ap + k);                  // K = k+2h, k+2h+1
        v2f bf;
        bf.x = wp[(k    ) * NCOLS];                     // K = k+2h   (coalesced)
        bf.y = wp[(k + 1) * NCOLS];                     // K = k+2h+1 (coalesced)

        // v_wmma_f32_16x16x4_f32: D = A(16x4) x B(4x16) + C
        acc = __builtin_amdgcn_wmma_f32_16x16x4_f32(
            /*neg_a=*/false, a, /*neg_b=*/false, bf,
            /*c_mod=*/(short)0, acc, /*reuse_a=*/false, /*reuse_b=*/false);
    }

    const float bb = colOK ? bias[col] : 0.0f;
    // C/D layout: VGPR r, lanes 0-15 -> row m0+r; lanes 16-31 -> row m0+r+8.
#pragma unroll
    for (int r = 0; r < 8; ++r) {
        const int row = m0 + r + 8 * h;
        X[row * NPAD + n0 + l] = colOK ? (acc[r] + bb) : 0.0f;
    }
}

// ---------------------------------------------------------------------------
// Kernel 2: per projection-kernel k: pairwise exp(-L1) sums over batch halves.
// 100 blocks x 256 threads (thread = sample i). 256x5 slice staged in LDS via
// gfx1250 async-to-LDS; j-loop reads are wave-uniform -> conflict-free.
// ---------------------------------------------------------------------------
__global__ __launch_bounds__(256) void pairwise_expl1(
    const float* __restrict__ X,    // 256 x 512 padded
    float*       __restrict__ out)  // 256 x 1224
{
    __shared__ float xs[B_SZ * 5];
    const int i  = threadIdx.x;
    const int kk = blockIdx.x;

#if USE_ASYNC_LDS
    {
        const float* src = X + i * NPAD + kk * 5;
#pragma unroll
        for (int d = 0; d < 5; ++d)
            __builtin_amdgcn_global_load_async_to_lds_b32(
                (glb_i32*)(src + d), (lds_i32*)&xs[i * 5 + d],
                /*imm offset*/0, /*cpol*/0);
        __builtin_amdgcn_s_wait_asynccnt(0);   // s_wait_asynccnt 0
    }
#else
#pragma unroll
    for (int d = 0; d < 5; ++d)
        xs[i * 5 + d] = X[i * NPAD + kk * 5 + d];
#endif
    __syncthreads();

    const float x0 = xs[i * 5 + 0];
    const float x1 = xs[i * 5 + 1];
    const float x2 = xs[i * 5 + 2];
    const float x3 = xs[i * 5 + 3];
    const float x4 = xs[i * 5 + 4];

    float num1 = 0.0f, num2 = 0.0f;

#pragma unroll 4
    for (int j = 0; j < 128; ++j) {
        float l1 = fabsf(x0 - xs[j * 5 + 0]) + fabsf(x1 - xs[j * 5 + 1])
                 + fabsf(x2 - xs[j * 5 + 2]) + fabsf(x3 - xs[j * 5 + 3])
                 + fabsf(x4 - xs[j * 5 + 4]);
        float s = __expf(-l1);          // v_exp_f32 (TRANS, co-executes)
        num1 += (j == i) ? 0.0f : s;    // mask self-pair
    }
#pragma unroll 4
    for (int j = 128; j < 256; ++j) {
        float l1 = fabsf(x0 - xs[j * 5 + 0]) + fabsf(x1 - xs[j * 5 + 1])
                 + fabsf(x2 - xs[j * 5 + 2]) + fabsf(x3 - xs[j * 5 + 3])
                 + fabsf(x4 - xs[j * 5 + 4]);
        float s = __expf(-l1);
        num2 += (j == i) ? 0.0f : s;
    }

    out[i * OUTW + 1024 + kk] = num1 * INV_DEN;
    out[i * OUTW + 1124 + kk] = num2 * INV_DEN;
}

// ---------------------------------------------------------------------------
// Kernel 3: out[:, :1024] = inputs, float4-wide (row stride 1224f is 16B
// aligned). 65536 float4s total.
// ---------------------------------------------------------------------------
__global__ __launch_bounds__(256) void copy_inputs(
    const float* __restrict__ in, float* __restrict__ out)
{
    const int t = blockIdx.x * 256 + threadIdx.x;  // 0..65535
    const int i = t >> 8;                           // row
    const int c = t & 255;                          // float4 col
    reinterpret_cast<float4*>(out + i * OUTW)[c] =
        reinterpret_cast<const float4*>(in + i * F_SZ)[c];
}

// ---------------------------------------------------------------------------
extern "C" void kernel_launch(void* const* d_in, const int* in_sizes, int n_in,
                              void* d_out, int out_size, void* d_ws, size_t ws_size,
                              hipStream_t stream) {
    const float* inputs = (const float*)d_in[0];   // 256*1024
    const float* W      = (const float*)d_in[1];   // 1024*500
    const float* bias   = (const float*)d_in[2];   // 500
    // d_in[3] = half (=128), baked into INV_DEN and loop splits.
    float* out = (float*)d_out;                    // 256*1224
    float* X   = (float*)d_ws;                     // 256*512 fp32 scratch

    // GEMM: 32 N-tiles (4 blocks x 8 waves) x 16 M-tiles.
    dim3 ggrid(4, 16);
    gemm_wmma_f32<<<ggrid, 256, 0, stream>>>(inputs, W, bias, X);

    // Copy inputs into out[:, :1024] (independent of GEMM).
    copy_inputs<<<256, 256, 0, stream>>>(inputs, out);

    // Pairwise exp(-L1) features (depends on X; same stream -> ordered).
    pairwise_expl1<<<100, 256, 0, stream>>>(X, out);
}